// GABlock_4355096838988
// MI455X (gfx1250) — compile-verified
//
#include <hip/hip_runtime.h>
#include <math.h>
#include <stdint.h>

// ---------------------------------------------------------------------------
// GABlock for MI455X (gfx1250, wave32).  All GEMM-like contractions lowered to
// v_wmma_f32_16x16x4_f32 (fp32 matches the fp32 reference; workload is HBM-
// bound on z (256MB)).  The big z streaming pass (feat_pair) uses CDNA5
// GLOBAL_LOAD_ASYNC_TO_LDS_B128 double-buffering (ASYNCcnt) to decouple the
// HBM stream from WMMA issue.
// ---------------------------------------------------------------------------

typedef float v2f __attribute__((ext_vector_type(2)));
typedef float v8f __attribute__((ext_vector_type(8)));

static constexpr int N  = 4;
static constexpr int L  = 512;
static constexpr int DN = 128;
static constexpr int DP = 64;
static constexpr int H  = 8;
static constexpr int HD = 32;
static constexpr int QKVD = H * HD;          // 256
static constexpr int AGGD = H * HD + H * DP; // 768

#define INV_SQRT_HD 0.17677669529663687f
#define SQRT_HALF   0.7071067811865476f
#define LN_EPS      1e-5f

// D = A(16x4) * B(4x16) + C   -- fp32 WMMA
__device__ __forceinline__ v8f wmma4(v2f a, v2f b, v8f c) {
  return __builtin_amdgcn_wmma_f32_16x16x4_f32(
      /*neg_a=*/false, a, /*neg_b=*/false, b,
      /*c_mod=*/(short)0, c, /*reuse_a=*/false, /*reuse_b=*/false);
}

__device__ __forceinline__ float wave_sum(float v) {
  #pragma unroll
  for (int m = 16; m >= 1; m >>= 1) v += __shfl_xor(v, m, 32);
  return v;
}
__device__ __forceinline__ float wave_max(float v) {
  #pragma unroll
  for (int m = 16; m >= 1; m >>= 1) v = fmaxf(v, __shfl_xor(v, m, 32));
  return v;
}

// ---------------------------------------------------------------------------
// K1: q/k/v projections.  x:(N*L,128) @ W:(128,256) -> (N*L,256) each.
// One 16x16 output tile per wave, K=128 (32 WMMAs).  6144 tiles / 8 waves.
// ---------------------------------------------------------------------------
__global__ __launch_bounds__(256) void k_qkv(
    const float* __restrict__ x,
    const float* __restrict__ Wq, const float* __restrict__ Wk,
    const float* __restrict__ Wv,
    float* __restrict__ q, float* __restrict__ kk, float* __restrict__ v) {
  const int lane = threadIdx.x & 31, wv = threadIdx.x >> 5;
  const int lr = lane & 15, lh = lane >> 4;
  const int tile = blockIdx.x * 8 + wv;       // 3 * 128 * 16 = 6144
  const int mat = tile >> 11;                 // which of q/k/v
  const int rem = tile & 2047;
  const int mt = rem >> 4, nt = rem & 15;
  const float* W = (mat == 0) ? Wq : (mat == 1) ? Wk : Wv;
  float*       O = (mat == 0) ? q  : (mat == 1) ? kk : v;
  const int row0 = mt * 16, col = nt * 16 + lr;

  v8f acc = {};
  #pragma unroll 4
  for (int t = 0; t < DN / 4; ++t) {
    const int kb = 4 * t + 2 * lh;
    v2f a, b;
    a.x = x[(row0 + lr) * DN + kb];
    a.y = x[(row0 + lr) * DN + kb + 1];
    b.x = W[kb * QKVD + col];
    b.y = W[(kb + 1) * QKVD + col];
    acc = wmma4(a, b, acc);
  }
  #pragma unroll
  for (int r = 0; r < 8; ++r)
    O[(row0 + r + 8 * lh) * QKVD + col] = acc[r];
}

// ---------------------------------------------------------------------------
// K2: logits (node + pair) + softmax -> alpha[n][h][i][j].
// One workgroup per (n, 16-row i tile).  Dynamic LDS holds 8 heads x 16 i x
// 512 j fp32 logits = 256KB (within the 320KB WGP LDS).
// ---------------------------------------------------------------------------
__global__ __launch_bounds__(256) void k_alpha(
    const float* __restrict__ q, const float* __restrict__ kk,
    const float* __restrict__ z, const float* __restrict__ Wp,
    float* __restrict__ alpha) {
  extern __shared__ float slog[];             // [H][16][L]
  const int lane = threadIdx.x & 31, wv = threadIdx.x >> 5;
  const int lr = lane & 15, lh = lane >> 4;
  const int n  = blockIdx.x >> 5;
  const int i0 = (blockIdx.x & 31) * 16;

  // ---- phase A: logits_node / sqrt(HD); wave wv == head h ----
  {
    const int h = wv;
    v2f aq[HD / 4];
    #pragma unroll
    for (int t = 0; t < HD / 4; ++t) {
      const int kb = 4 * t + 2 * lh;
      aq[t].x = q[(n * L + i0 + lr) * QKVD + h * HD + kb];
      aq[t].y = q[(n * L + i0 + lr) * QKVD + h * HD + kb + 1];
    }
    for (int jt = 0; jt < L / 16; ++jt) {
      v8f acc = {};
      #pragma unroll
      for (int t = 0; t < HD / 4; ++t) {
        const int kb = 4 * t + 2 * lh;
        v2f b;
        b.x = kk[(n * L + jt * 16 + lr) * QKVD + h * HD + kb];
        b.y = kk[(n * L + jt * 16 + lr) * QKVD + h * HD + kb + 1];
        acc = wmma4(aq[t], b, acc);
      }
      #pragma unroll
      for (int r = 0; r < 8; ++r)
        slog[(h * 16 + r + 8 * lh) * L + jt * 16 + lr] = acc[r] * INV_SQRT_HD;
    }
  }
  __syncthreads();

  // ---- phase B: += z @ Wp.  Tile = (i, jtile); A = 16 z rows (M=j, K=p),
  // B = Wp padded to 64x16 (cols 8..15 zero).  C: M=j_local, N=head. ----
  for (int tt = wv; tt < 16 * (L / 16); tt += 8) {
    const int i = tt >> 5, jt = tt & 31;
    const float* zrow = z + ((size_t)(n * L + i0 + i) * L + jt * 16) * DP;
    v8f acc = {};
    #pragma unroll
    for (int t = 0; t < DP / 4; ++t) {
      const int kb = 4 * t + 2 * lh;
      __builtin_prefetch(zrow + lr * DP + kb + 16, 0, 1);
      v2f a, b;
      a.x = zrow[lr * DP + kb];
      a.y = zrow[lr * DP + kb + 1];
      b.x = (lr < H) ? Wp[kb * H + lr] : 0.0f;
      b.y = (lr < H) ? Wp[(kb + 1) * H + lr] : 0.0f;
      acc = wmma4(a, b, acc);
    }
    if (lr < H) {
      #pragma unroll
      for (int r = 0; r < 8; ++r)
        slog[(lr * 16 + i) * L + jt * 16 + r + 8 * lh] += acc[r];
    }
  }
  __syncthreads();

  // ---- phase C: softmax over j (apply sqrt(0.5)); 16 rows per wave ----
  for (int rr = wv * 16; rr < wv * 16 + 16; ++rr) {
    const int h = rr >> 4, i = rr & 15;
    const float* rowp = slog + (size_t)rr * L;
    float vals[L / 32];
    float m = -1e30f;
    #pragma unroll
    for (int u = 0; u < L / 32; ++u) {
      vals[u] = rowp[lane + 32 * u] * SQRT_HALF;
      m = fmaxf(m, vals[u]);
    }
    m = wave_max(m);
    float s = 0.f;
    #pragma unroll
    for (int u = 0; u < L / 32; ++u) {
      vals[u] = __expf(vals[u] - m);
      s += vals[u];
    }
    s = wave_sum(s);
    const float inv = 1.0f / s;
    float* ap = alpha + ((size_t)(n * H + h) * L + i0 + i) * L;
    #pragma unroll
    for (int u = 0; u < L / 32; ++u)
      ap[lane + 32 * u] = vals[u] * inv;
  }
}

// ---------------------------------------------------------------------------
// K3: feat_node = alpha @ v.  Per (n,h,itile): (16x512)@(512x32), K=512.
// ---------------------------------------------------------------------------
__global__ __launch_bounds__(256) void k_featnode(
    const float* __restrict__ alpha, const float* __restrict__ v,
    float* __restrict__ agg) {
  const int lane = threadIdx.x & 31, wv = threadIdx.x >> 5;
  const int lr = lane & 15, lh = lane >> 4;
  const int id = blockIdx.x * 8 + wv;          // 1024 tiles
  const int n = id >> 8, h = (id >> 5) & 7, it = id & 31;
  const int i0 = it * 16;
  const float* ap = alpha + ((size_t)(n * H + h) * L + i0) * L;

  v8f acc0 = {}, acc1 = {};
  #pragma unroll 2
  for (int t = 0; t < L / 4; ++t) {
    const int kb = 4 * t + 2 * lh;
    v2f a, b0, b1;
    a.x = ap[lr * L + kb];
    a.y = ap[lr * L + kb + 1];
    const float* v0 = v + (size_t)(n * L + kb) * QKVD + h * HD;
    b0.x = v0[lr];        b0.y = v0[QKVD + lr];
    b1.x = v0[16 + lr];   b1.y = v0[QKVD + 16 + lr];
    acc0 = wmma4(a, b0, acc0);
    acc1 = wmma4(a, b1, acc1);
  }
  #pragma unroll
  for (int r = 0; r < 8; ++r) {
    float* o = agg + (size_t)(n * L + i0 + r + 8 * lh) * AGGD + h * HD;
    o[lr]      = acc0[r];
    o[16 + lr] = acc1[r];
  }
}

// ---------------------------------------------------------------------------
// K4: feat_pair = alpha^T @ z_i.  Per (n,i): (8pad16 x 512)@(512x64), K=512.
// The z_i stream (128KB/wave, 256MB total = the dominant HBM traffic of the
// whole op) is staged through LDS with double-buffered
// GLOBAL_LOAD_ASYNC_TO_LDS_B128 (ASYNCcnt): each 16jx64p tile (4KB) = 8 async
// b128 ops (512B each); s_wait_asynccnt 8 overlaps the next tile's fill with
// this tile's WMMAs.  A rows 8..15 zero-padded; C rows 8..15 not stored.
// ---------------------------------------------------------------------------
__global__ __launch_bounds__(256) void k_featpair(
    const float* __restrict__ alpha, const float* __restrict__ z,
    float* __restrict__ agg) {
  __shared__ __align__(16) float zstage[8][2][16 * DP];  // 8 waves x 2 x 4KB
  const int lane = threadIdx.x & 31, wv = threadIdx.x >> 5;
  const int lr = lane & 15, lh = lane >> 4;
  const int id = blockIdx.x * 8 + wv;          // 2048 tiles
  const int n = id >> 9, i = id & 511;
  const float* zi = z + (size_t)(n * L + i) * L * DP;
  const float* ap = alpha + ((size_t)(n * H) * L + i) * L;   // +h*L*L per head

  const uint64_t zi_u64 = (uint64_t)(uintptr_t)zi;
  const uint32_t lds0 = (uint32_t)(uintptr_t)&zstage[wv][0][0];
  const uint32_t lds1 = (uint32_t)(uintptr_t)&zstage[wv][1][0];
  const uint32_t l16 = (uint32_t)lane * 16u;

  auto issue_tile = [&](int tile, uint32_t ldsb) {
    const uint64_t ga = zi_u64 + (uint64_t)((uint32_t)tile * 4096u + l16);
    const uint32_t la = ldsb + l16;
    #pragma unroll
    for (int qq = 0; qq < 8; ++qq)
      asm volatile("global_load_async_to_lds_b128 %0, %1, off offset:%2"
                   :: "v"(la), "v"(ga), "i"(qq * 512) : "memory");
  };

  issue_tile(0, lds0);
  issue_tile(1, lds1);

  v8f acc[4] = {v8f{}, v8f{}, v8f{}, v8f{}};
  for (int tile = 0; tile < L / 16; ++tile) {
    if (tile < L / 16 - 1)
      asm volatile("s_wait_asynccnt 0x8" ::: "memory");
    else
      asm volatile("s_wait_asynccnt 0x0" ::: "memory");
    const float* sb = &zstage[wv][tile & 1][0];
    #pragma unroll
    for (int tl = 0; tl < 4; ++tl) {
      const int kb = 4 * tl + 2 * lh;        // local j within tile
      const int kg = tile * 16 + kb;         // global j
      v2f a;
      a.x = (lr < H) ? ap[(size_t)lr * L * L + kg] : 0.0f;
      a.y = (lr < H) ? ap[(size_t)lr * L * L + kg + 1] : 0.0f;
      #pragma unroll
      for (int pt = 0; pt < 4; ++pt) {
        v2f b;
        b.x = sb[kb * DP + pt * 16 + lr];
        b.y = sb[(kb + 1) * DP + pt * 16 + lr];
        acc[pt] = wmma4(a, b, acc[pt]);
      }
    }
    if (tile + 2 < L / 16)
      issue_tile(tile + 2, (tile & 1) ? lds1 : lds0);
  }

  if (lh == 0) {                                // C rows 0..7 == heads
    #pragma unroll
    for (int r = 0; r < 8; ++r) {
      float* o = agg + (size_t)(n * L + i) * AGGD + QKVD + r * DP;
      #pragma unroll
      for (int pt = 0; pt < 4; ++pt) o[pt * 16 + lr] = acc[pt][r];
    }
  }
}

// ---------------------------------------------------------------------------
// K5: fused transition.  Per (n, itile):
//   t1 = x + agg@Wt1 + bt1 ; feats = LN1(t1)
//   h1 = relu(feats@W2a + b2a) ; o = feats + h1@W2b + b2b ; out = LN2(o)
// Wave wv owns output columns [16wv, 16wv+16); LN stats via wave32 shuffles.
// ---------------------------------------------------------------------------
__global__ __launch_bounds__(256) void k_out(
    const float* __restrict__ x,   const float* __restrict__ agg,
    const float* __restrict__ Wt1, const float* __restrict__ bt1,
    const float* __restrict__ g1,  const float* __restrict__ b1,
    const float* __restrict__ W2a, const float* __restrict__ b2a,
    const float* __restrict__ W2b, const float* __restrict__ b2b,
    const float* __restrict__ g2,  const float* __restrict__ b2,
    float* __restrict__ out) {
  __shared__ float feats[16][DN];
  __shared__ float buf[16][DN];
  const int lane = threadIdx.x & 31, wv = threadIdx.x >> 5;
  const int lr = lane & 15, lh = lane >> 4;
  const int n  = blockIdx.x >> 5;
  const int i0 = (blockIdx.x & 31) * 16;
  const int col = wv * 16 + lr;

  // ---- stage 1: t1 = x + agg @ Wt1 + bt1, K = 768 ----
  {
    v8f acc = {};
    #pragma unroll 4
    for (int t = 0; t < AGGD / 4; ++t) {
      const int kb = 4 * t + 2 * lh;
      v2f a, b;
      a.x = agg[(size_t)(n * L + i0 + lr) * AGGD + kb];
      a.y = agg[(size_t)(n * L + i0 + lr) * AGGD + kb + 1];
      b.x = Wt1[kb * DN + col];
      b.y = Wt1[(kb + 1) * DN + col];
      acc = wmma4(a, b, acc);
    }
    #pragma unroll
    for (int r = 0; r < 8; ++r) {
      const int m = r + 8 * lh;
      buf[m][col] = acc[r] + x[(size_t)(n * L + i0 + m) * DN + col] + bt1[col];
    }
  }
  __syncthreads();

  // ---- LN1: wave wv owns rows 2wv, 2wv+1 ----
  #pragma unroll
  for (int rr = 0; rr < 2; ++rr) {
    const int row = wv * 2 + rr;
    float e[DN / 32], s = 0.f, s2 = 0.f;
    #pragma unroll
    for (int u = 0; u < DN / 32; ++u) {
      e[u] = buf[row][lane + 32 * u];
      s += e[u]; s2 += e[u] * e[u];
    }
    s = wave_sum(s); s2 = wave_sum(s2);
    const float mean = s / DN;
    const float rstd = rsqrtf(s2 / DN - mean * mean + LN_EPS);
    #pragma unroll
    for (int u = 0; u < DN / 32; ++u) {
      const int c = lane + 32 * u;
      feats[row][c] = (e[u] - mean) * rstd * g1[c] + b1[c];
    }
  }
  __syncthreads();

  // ---- stage 2: h1 = relu(feats @ W2a + b2a), K = 128 ----
  {
    v8f acc = {};
    #pragma unroll
    for (int t = 0; t < DN / 4; ++t) {
      const int kb = 4 * t + 2 * lh;
      v2f a, b;
      a.x = feats[lr][kb]; a.y = feats[lr][kb + 1];
      b.x = W2a[kb * DN + col]; b.y = W2a[(kb + 1) * DN + col];
      acc = wmma4(a, b, acc);
    }
    __syncthreads();
    #pragma unroll
    for (int r = 0; r < 8; ++r)
      buf[r + 8 * lh][col] = fmaxf(acc[r] + b2a[col], 0.f);
  }
  __syncthreads();

  // ---- stage 3: o = feats + h1 @ W2b + b2b, K = 128 ----
  v8f acc3 = {};
  #pragma unroll
  for (int t = 0; t < DN / 4; ++t) {
    const int kb = 4 * t + 2 * lh;
    v2f a, b;
    a.x = buf[lr][kb]; a.y = buf[lr][kb + 1];
    b.x = W2b[kb * DN + col]; b.y = W2b[(kb + 1) * DN + col];
    acc3 = wmma4(a, b, acc3);
  }
  __syncthreads();
  #pragma unroll
  for (int r = 0; r < 8; ++r) {
    const int m = r + 8 * lh;
    buf[m][col] = acc3[r] + b2b[col] + feats[m][col];
  }
  __syncthreads();

  // ---- LN2 + store ----
  #pragma unroll
  for (int rr = 0; rr < 2; ++rr) {
    const int row = wv * 2 + rr;
    float e[DN / 32], s = 0.f, s2 = 0.f;
    #pragma unroll
    for (int u = 0; u < DN / 32; ++u) {
      e[u] = buf[row][lane + 32 * u];
      s += e[u]; s2 += e[u] * e[u];
    }
    s = wave_sum(s); s2 = wave_sum(s2);
    const float mean = s / DN;
    const float rstd = rsqrtf(s2 / DN - mean * mean + LN_EPS);
    #pragma unroll
    for (int u = 0; u < DN / 32; ++u) {
      const int c = lane + 32 * u;
      out[(size_t)(n * L + i0 + row) * DN + c] =
          (e[u] - mean) * rstd * g2[c] + b2[c];
    }
  }
}

// ---------------------------------------------------------------------------
extern "C" void kernel_launch(void* const* d_in, const int* in_sizes, int n_in,
                              void* d_out, int out_size, void* d_ws,
                              size_t ws_size, hipStream_t stream) {
  const float* x   = (const float*)d_in[0];
  const float* z   = (const float*)d_in[1];
  const float* Wq  = (const float*)d_in[2];
  const float* Wk  = (const float*)d_in[3];
  const float* Wv  = (const float*)d_in[4];
  const float* Wp  = (const float*)d_in[5];
  const float* Wt1 = (const float*)d_in[6];
  const float* bt1 = (const float*)d_in[7];
  const float* g1  = (const float*)d_in[8];
  const float* b1  = (const float*)d_in[9];
  const float* W2a = (const float*)d_in[10];
  const float* b2a = (const float*)d_in[11];
  const float* W2b = (const float*)d_in[12];
  const float* b2b = (const float*)d_in[13];
  const float* g2  = (const float*)d_in[14];
  const float* b2  = (const float*)d_in[15];

  float* ws    = (float*)d_ws;
  float* q     = ws;                                     // N*L*256
  float* k     = q + (size_t)N * L * QKVD;
  float* v     = k + (size_t)N * L * QKVD;
  float* alpha = v + (size_t)N * L * QKVD;               // N*H*L*L
  float* agg   = alpha + (size_t)N * H * L * L;          // N*L*768
  float* out   = (float*)d_out;

  k_qkv<<<768, 256, 0, stream>>>(x, Wq, Wk, Wv, q, k, v);
  k_alpha<<<N * (L / 16), 256, (size_t)H * 16 * L * sizeof(float), stream>>>(
      q, k, z, Wp, alpha);
  k_featnode<<<(N * H * (L / 16)) / 8, 256, 0, stream>>>(alpha, v, agg);
  k_featpair<<<(N * L) / 8, 256, 0, stream>>>(alpha, z, agg);
  k_out<<<N * (L / 16), 256, 0, stream>>>(x, agg, Wt1, bt1, g1, b1,
                                          W2a, b2a, W2b, b2b, g2, b2, out);
}